// MeshIntersection_42872363549126
// MI455X (gfx1250) — compile-verified
//
#include <hip/hip_runtime.h>

typedef __attribute__((ext_vector_type(2))) float v2f;
typedef __attribute__((ext_vector_type(8))) float v8f;

#define MESH_EPS 1e-8f
#define F_TRIS   2048

// ---------------------------------------------------------------------------
// Kernel 1: per-triangle precompute.
// For ray dir d=(0,0,1), Moller-Trumbore quantities are affine in the point p:
//   u = f*dot(p-v1, h)            h = (-e2y, e2x, 0)
//   v = f*((p-v1)x*e1y - (p-v1)y*e1x)
//   t = f*dot(p-v1, e1 x e2)
// Store each as float4 (cx, cy, cz, bias) so that  q = c . p + bias.
// Degenerate triangles get f=0 -> u=v=t=0 -> t>EPS fails -> no hit, matching
// the reference's valid-mask semantics.
// Workspace layout (single contiguous block so the main loop can use one base
// pointer + immediate offsets):  qU[2048] | qV[2048] | qT[2048]  (float4).
// ---------------------------------------------------------------------------
__global__ void mesh_precompute(const float* __restrict__ V1,
                                const float* __restrict__ V2,
                                const float* __restrict__ V3,
                                float4* __restrict__ qU) {
    int f = blockIdx.x * blockDim.x + threadIdx.x;
    if (f >= F_TRIS) return;
    float4* qV = qU + F_TRIS;
    float4* qT = qV + F_TRIS;

    float v1x = V1[3*f+0], v1y = V1[3*f+1], v1z = V1[3*f+2];
    float e1x = V2[3*f+0]-v1x, e1y = V2[3*f+1]-v1y, e1z = V2[3*f+2]-v1z;
    float e2x = V3[3*f+0]-v1x, e2y = V3[3*f+1]-v1y, e2z = V3[3*f+2]-v1z;

    float a  = -e1x*e2y + e1y*e2x;        // dot(e1, h)  (determinant)
    bool  ok = fabsf(a) > MESH_EPS;
    float fi = ok ? 1.0f/a : 0.0f;

    // u coefficients
    float ux = -fi*e2y, uy = fi*e2x;
    float ub = -(ux*v1x + uy*v1y);
    // v coefficients
    float vx = fi*e1y, vy = -fi*e1x;
    float vb = -(vx*v1x + vy*v1y);
    // t coefficients: fi * (e1 x e2)
    float nx = e1y*e2z - e1z*e2y;
    float ny = e1z*e2x - e1x*e2z;
    float nz = e1x*e2y - e1y*e2x;
    float tx = fi*nx, ty = fi*ny, tz = fi*nz;
    float tb = -(tx*v1x + ty*v1y + tz*v1z);

    qU[f] = make_float4(ux, uy, 0.0f, ub);
    qV[f] = make_float4(vx, vy, 0.0f, vb);
    qT[f] = make_float4(tx, ty, tz, tb);
}

// ---------------------------------------------------------------------------
// Kernel 2: WMMA intersection counting.
// One wave per 16-point tile. A operand (16x4 f32, [x y z 1] rows) loaded
// once; loop over 128 triangle tiles: 3x V_WMMA_F32_16X16X4_F32 (u,v,t for
// 16 points x 16 triangles), then a min3-collapsed hit test:
//   hit = (min3(u, v, 1-u-v) >= 0) && (t > EPS)        [u<=1 is implied]
// C/D layout: VGPR r, lane l -> point M = r + 8*(l>=16), triangle N = l&15.
// Counts reduce over N with 16-lane-half xor shuffles.
// ---------------------------------------------------------------------------
__global__ void __launch_bounds__(256)
mesh_intersect_wmma(const float*  __restrict__ pts,
                    const float2* __restrict__ coef,   // base of qU|qV|qT
                    float* __restrict__ out) {
    const int lane  = threadIdx.x & 31;
    const int wave  = threadIdx.x >> 5;
    const int tile  = blockIdx.x * (blockDim.x >> 5) + wave;
    const int pbase = tile * 16;

    // --- A operand: 16x4 matrix. lane<16: (K0,K1)=(px,py) of point M=lane.
    //                             lane>=16: (K2,K3)=(pz,1.0) of point M=lane-16.
    v2f aop;
    {
        int p = pbase + (lane & 15);
        if (lane < 16) {
            aop.x = pts[3*p + 0];
            aop.y = pts[3*p + 1];
        } else {
            aop.x = pts[3*p + 2];
            aop.y = 1.0f;
        }
    }

    const int half = lane >> 4;            // 0: rows K0..1, 1: rows K2..3
    const int tl   = lane & 15;            // triangle column within tile
    const int QV_OFF = F_TRIS * 2;         // float2 offset to qV block (32 KB)
    const int QT_OFF = F_TRIS * 4;         // float2 offset to qT block (64 KB)

    v8f acc = {0.f,0.f,0.f,0.f,0.f,0.f,0.f,0.f};
    const v8f czero = {0.f,0.f,0.f,0.f,0.f,0.f,0.f,0.f};

    for (int tt = 0; tt < F_TRIS/16; ++tt) {
        const int idx = 2*(tt*16 + tl) + half;
        float2 bu2 = coef[idx];                 // one base reg pair; qV/qT hit
        float2 bv2 = coef[idx + QV_OFF];        // via 24-bit immediate offsets
        float2 bt2 = coef[idx + QT_OFF];
        v2f bu; bu.x = bu2.x; bu.y = bu2.y;
        v2f bv; bv.x = bv2.x; bv.y = bv2.y;
        v2f bt; bt.x = bt2.x; bt.y = bt2.y;

        v8f U = __builtin_amdgcn_wmma_f32_16x16x4_f32(false, aop, false, bu,
                                                      (short)0, czero, false, false);
        v8f V = __builtin_amdgcn_wmma_f32_16x16x4_f32(false, aop, false, bv,
                                                      (short)0, czero, false, false);
        v8f T = __builtin_amdgcn_wmma_f32_16x16x4_f32(false, aop, false, bt,
                                                      (short)0, czero, false, false);
        #pragma unroll
        for (int r = 0; r < 8; ++r) {
            float uu = U[r], vv = V[r], tv = T[r];
            float w  = 1.0f - uu - vv;
            float mn = fminf(fminf(uu, vv), w);       // -> v_min3_f32
            bool hit = (mn >= 0.0f) & (tv > MESH_EPS);
            acc[r] += hit ? 1.0f : 0.0f;
        }
    }

    // Sum counts over the 16 triangle columns (xor masks < 16 stay in-half).
    #pragma unroll
    for (int r = 0; r < 8; ++r) {
        float s = acc[r];
        s += __shfl_xor(s, 1, 32);
        s += __shfl_xor(s, 2, 32);
        s += __shfl_xor(s, 4, 32);
        s += __shfl_xor(s, 8, 32);
        acc[r] = s;
    }
    if (tl == 0) {                       // lanes 0 and 16 write 8 points each
        const int pb = pbase + 8*half;
        #pragma unroll
        for (int r = 0; r < 8; ++r) out[pb + r] = acc[r];
    }
}

// ---------------------------------------------------------------------------
// Launch: precompute coefficients into d_ws, then WMMA intersection kernel.
// ---------------------------------------------------------------------------
extern "C" void kernel_launch(void* const* d_in, const int* in_sizes, int n_in,
                              void* d_out, int out_size, void* d_ws, size_t ws_size,
                              hipStream_t stream) {
    const float* pts = (const float*)d_in[0];   // [4,16384,3]
    const float* V1  = (const float*)d_in[1];   // [2048,3]
    const float* V2  = (const float*)d_in[2];
    const float* V3  = (const float*)d_in[3];

    float4* qU = (float4*)d_ws;                 // qU|qV|qT contiguous, 96 KB

    mesh_precompute<<<(F_TRIS + 255)/256, 256, 0, stream>>>(V1, V2, V3, qU);

    const int npts  = in_sizes[0] / 3;          // 65536
    const int waves = npts / 16;                // 4096 point-tiles
    const int wpb   = 8;                        // waves per block (256 threads)
    mesh_intersect_wmma<<<waves / wpb, wpb * 32, 0, stream>>>(
        pts, (const float2*)qU, (float*)d_out);
}